// SqueezeAndExcitation_30210799960604
// MI455X (gfx1250) — compile-verified
//
#include <hip/hip_runtime.h>
#include <hip/hip_bf16.h>
#include <math.h>

// Sizes fixed by the reference
#define BB   1024
#define CC   256
#define HW   361        // 19*19
#define SE_  64
#define K1   768        // 3*C
#define N2   512        // 2*C
#define B_AVG 14.0f     // (19+9)/2

typedef __attribute__((ext_vector_type(2))) float v2f;
typedef __attribute__((ext_vector_type(8))) float v8f;

// ---------------------------------------------------------------------------
// Kernel 1: masked mean/max pooling. One wave32 per (b,c) channel.
// pooled[b, 0:256]   = mean
// pooled[b, 256:512] = mean * (sqrt(masksum)-B_AVG)/10
// pooled[b, 512:768] = masked max
// ---------------------------------------------------------------------------
__global__ __launch_bounds__(256) void pool_kernel(
    const float* __restrict__ x, const float* __restrict__ mask,
    const float* __restrict__ msum, const float* __restrict__ msqrt,
    float* __restrict__ pooled) {
  const int wid  = blockIdx.x * (blockDim.x >> 5) + (threadIdx.x >> 5);
  const int lane = threadIdx.x & 31;
  const int bc   = wid;                 // 0 .. B*C-1
  const int b    = bc >> 8;             // C == 256
  const int c    = bc & 255;

  const float* xc = x    + (size_t)bc * HW;
  const float* mc = mask + (size_t)b  * HW;

  float s  = 0.0f;
  float mx = -3.4e38f;
  for (int i = lane; i < HW; i += 32) {
    const float v = xc[i];
    const float m = mc[i];
    s += v;                                      // reference sums raw x
    mx = fmaxf(mx, v + (1.0f - m) * -5000.0f);   // CRAZY_NEG masking
  }
  // wave32 butterfly reduction
  for (int off = 16; off > 0; off >>= 1) {
    s  += __shfl_xor(s, off, 32);
    mx  = fmaxf(mx, __shfl_xor(mx, off, 32));
  }
  if (lane == 0) {
    const float mean = s / msum[b];
    const float bd   = (msqrt[b] - B_AVG) * 0.1f;
    float* pr = pooled + (size_t)b * K1;
    pr[c]            = mean;
    pr[256 + c]      = mean * bd;
    pr[512 + c]      = mx;
  }
}

// ---------------------------------------------------------------------------
// Kernel 2: SE MLP with fp32 WMMA (V_WMMA_F32_16X16X4_F32).
// Grid: B/16 = 64 workgroups, 128 threads (4 waves) each.
// GEMM1: hid(16x64)  = relu(pooled(16x768) @ w1^T + b1)   [1 N-tile/wave]
// GEMM2: se (16x512) = hid @ w2^T + b2                    [8 N-tiles/wave]
// gamma half gets sigmoid() applied before store.
// ---------------------------------------------------------------------------
__global__ __launch_bounds__(128) void se_gemm_kernel(
    const float* __restrict__ pooled,
    const float* __restrict__ w1, const float* __restrict__ b1,
    const float* __restrict__ w2, const float* __restrict__ b2,
    float* __restrict__ gsig, float* __restrict__ betaw) {
  __shared__ float sp[16 * K1];   // 48 KB pooled tile
  __shared__ float sh[16 * SE_];  //  4 KB hidden tile

  const int t    = threadIdx.x;
  const int wave = t >> 5;
  const int lane = t & 31;
  const int m0   = blockIdx.x * 16;

  // Stage pooled tile (16 consecutive rows -> flat contiguous copy)
  for (int i = t; i < 16 * K1; i += 128) sp[i] = pooled[(size_t)m0 * K1 + i];
  __syncthreads();

  const int rc    = lane & 15;          // A-row (M) == B-col (N) index per lane
  const int kb    = (lane >> 4) << 1;   // K sub-offset: 0 for lanes 0-15, 2 for 16-31
  const int rbase = (lane >> 4) << 3;   // C/D rows: r (+8 for upper half-wave)

  // ---- GEMM1: each wave computes one 16-wide N-tile of hid ----
  {
    const int n0 = wave * 16;
    v8f acc = {};
    const float* arow = sp + rc * K1;
    const float* brow = w1 + (size_t)(n0 + rc) * K1;  // B[k][n] = w1[n][k]
    for (int k0 = 0; k0 < K1; k0 += 4) {
      v2f a, bf;
      a.x  = arow[k0 + kb];
      a.y  = arow[k0 + kb + 1];
      bf.x = brow[k0 + kb];
      bf.y = brow[k0 + kb + 1];
      acc = __builtin_amdgcn_wmma_f32_16x16x4_f32(
          false, a, false, bf, (short)0, acc, false, false);
    }
    const float bias = b1[n0 + rc];
    #pragma unroll
    for (int r = 0; r < 8; ++r) {
      float v = acc[r] + bias;
      v = v > 0.0f ? v : 0.0f;          // relu
      sh[(rbase + r) * SE_ + n0 + rc] = v;
    }
  }
  __syncthreads();

  // ---- GEMM2: each wave computes 8 16-wide N-tiles of se ----
  #pragma unroll
  for (int nt = 0; nt < 8; ++nt) {
    const int n0 = (wave * 8 + nt) * 16;              // 0..511, 256-aligned split
    v8f acc = {};
    const float* arow = sh + rc * SE_;
    const float* brow = w2 + (size_t)(n0 + rc) * SE_; // B[k][n] = w2[n][k]
    #pragma unroll
    for (int k0 = 0; k0 < SE_; k0 += 4) {
      v2f a, bf;
      a.x  = arow[k0 + kb];
      a.y  = arow[k0 + kb + 1];
      bf.x = brow[k0 + kb];
      bf.y = brow[k0 + kb + 1];
      acc = __builtin_amdgcn_wmma_f32_16x16x4_f32(
          false, a, false, bf, (short)0, acc, false, false);
    }
    const float bias = b2[n0 + rc];
    #pragma unroll
    for (int r = 0; r < 8; ++r) {
      const float v  = acc[r] + bias;
      const int   gm = m0 + rbase + r;
      if (n0 < 256) {  // gamma half: store sigmoid(gamma)
        gsig[(size_t)gm * CC + n0 + rc] = 1.0f / (1.0f + expf(-v));
      } else {         // beta half
        betaw[(size_t)gm * CC + (n0 - 256) + rc] = v;
      }
    }
  }
}

// ---------------------------------------------------------------------------
// Kernel 3: out = (sigmoid(gamma)*x + beta) * mask. One wave per (b,c).
// x is never re-read and out never read -> non-temporal hints.
// ---------------------------------------------------------------------------
__global__ __launch_bounds__(256) void apply_kernel(
    const float* __restrict__ x, const float* __restrict__ mask,
    const float* __restrict__ gsig, const float* __restrict__ betaw,
    float* __restrict__ out) {
  const int wid  = blockIdx.x * (blockDim.x >> 5) + (threadIdx.x >> 5);
  const int lane = threadIdx.x & 31;
  const int bc   = wid;
  const int b    = bc >> 8;

  const float g  = gsig[bc];
  const float be = betaw[bc];

  const float* xc = x    + (size_t)bc * HW;
  const float* mc = mask + (size_t)b  * HW;
  float*       oc = out  + (size_t)bc * HW;

  for (int i = lane; i < HW; i += 32) {
    const float v = __builtin_nontemporal_load(xc + i);
    const float m = mc[i];
    __builtin_nontemporal_store((g * v + be) * m, oc + i);
  }
}

// ---------------------------------------------------------------------------
extern "C" void kernel_launch(void* const* d_in, const int* in_sizes, int n_in,
                              void* d_out, int out_size, void* d_ws, size_t ws_size,
                              hipStream_t stream) {
  const float* x     = (const float*)d_in[0];
  const float* mask  = (const float*)d_in[1];
  const float* msum  = (const float*)d_in[2];
  const float* msqrt = (const float*)d_in[3];
  const float* w1    = (const float*)d_in[4];
  const float* b1    = (const float*)d_in[5];
  const float* w2    = (const float*)d_in[6];
  const float* b2    = (const float*)d_in[7];
  float* out = (float*)d_out;

  // Workspace layout (floats): pooled[B*768] | gsig[B*256] | beta[B*256]  = 5 MB
  float* pooled = (float*)d_ws;
  float* gsig   = pooled + (size_t)BB * K1;
  float* betaw  = gsig   + (size_t)BB * CC;

  // 1) pooling: B*C = 262144 channels, 8 waves (256 thr) per block
  pool_kernel<<<(BB * CC) / 8, 256, 0, stream>>>(x, mask, msum, msqrt, pooled);

  // 2) SE MLP via fp32 WMMA: 64 workgroups x 4 waves
  se_gemm_kernel<<<BB / 16, 128, 0, stream>>>(pooled, w1, b1, w2, b2, gsig, betaw);

  // 3) apply
  apply_kernel<<<(BB * CC) / 8, 256, 0, stream>>>(x, mask, gsig, betaw, out);
}